// WeightGenerator_V2_807453852207
// MI455X (gfx1250) — compile-verified
//
#include <hip/hip_runtime.h>

typedef float v2f __attribute__((ext_vector_type(2)));
typedef float v8f __attribute__((ext_vector_type(8)));
typedef unsigned int u32x4 __attribute__((ext_vector_type(4)));
typedef int i32x8 __attribute__((ext_vector_type(8)));
typedef int i32x4 __attribute__((ext_vector_type(4)));

#define BN_EPS 1e-5f

// ---------------------------------------------------------------------------
// Kernel A: 1x1 conv  t[b][ch][pos] = sum_c x[b][c][pos] * conv1_w[ch][c] + b
// ---------------------------------------------------------------------------
__global__ __launch_bounds__(128) void k_conv1(const float* __restrict__ x,
                                               const float* __restrict__ w,
                                               const float* __restrict__ bias,
                                               float* __restrict__ t) {
  const int pos = blockIdx.x;   // 0..255  (i*16+j)
  const int b   = blockIdx.y;   // 0..1
  const int tid = threadIdx.x;  // 0..127
  __shared__ float xs[256];
  xs[tid]       = x[((size_t)(b * 256 + tid)) * 256 + pos];
  xs[tid + 128] = x[((size_t)(b * 256 + tid + 128)) * 256 + pos];
  __syncthreads();
  const float4* wr = (const float4*)(w + (size_t)tid * 256);
  float s = bias[tid];
#pragma unroll 8
  for (int c4 = 0; c4 < 64; ++c4) {
    float4 wv = wr[c4];
    s = fmaf(wv.x, xs[c4 * 4 + 0], s);
    s = fmaf(wv.y, xs[c4 * 4 + 1], s);
    s = fmaf(wv.z, xs[c4 * 4 + 2], s);
    s = fmaf(wv.w, xs[c4 * 4 + 3], s);
  }
  t[((size_t)(b * 128 + tid)) * 256 + pos] = s;
}

// ---------------------------------------------------------------------------
// Kernel B: depthwise 3x3 + BN + StarReLU + channel-expand + StarReLU -> wdec
// ---------------------------------------------------------------------------
__global__ __launch_bounds__(128) void k_dw_ce(
    const float* __restrict__ t, const float* __restrict__ dw_w,
    const float* __restrict__ dw_b, const float* __restrict__ gamma,
    const float* __restrict__ beta, const float* __restrict__ mean,
    const float* __restrict__ var, const float* __restrict__ sr1s,
    const float* __restrict__ sr1b, const float* __restrict__ ce_w,
    const float* __restrict__ ce_b, const float* __restrict__ sr2s,
    const float* __restrict__ sr2b, float* __restrict__ wdec) {
  const int pos = blockIdx.x;
  const int b   = blockIdx.y;
  const int i   = pos >> 4;
  const int j   = pos & 15;
  const int ch  = threadIdx.x;  // 0..127

  const float* trow = t + ((size_t)(b * 128 + ch)) * 256;
  float acc = dw_b[ch];
#pragma unroll
  for (int dy = 0; dy < 3; ++dy) {
#pragma unroll
    for (int dx = 0; dx < 3; ++dx) {
      int ii = i + dy - 1, jj = j + dx - 1;
      if (ii >= 0 && ii < 16 && jj >= 0 && jj < 16)
        acc = fmaf(trow[ii * 16 + jj], dw_w[ch * 9 + dy * 3 + dx], acc);
    }
  }
  float v   = fmaf(gamma[ch] * rsqrtf(var[ch] + BN_EPS), acc - mean[ch], beta[ch]);
  float rl  = fmaxf(v, 0.0f);
  float s1  = fmaf(sr1s[0], rl * rl, sr1b[0]);

  __shared__ float ls[128];
  ls[ch] = s1;
  __syncthreads();

  if (ch < 8) {
    float a = ce_b[ch];
    const float* cw = ce_w + ch * 128;
#pragma unroll 4
    for (int c = 0; c < 128; ++c) a = fmaf(cw[c], ls[c], a);
    float r = fmaxf(a, 0.0f);
    wdec[(size_t)b * 2048 + ch * 256 + pos] = fmaf(sr2s[0], r * r, sr2b[0]);
  }
}

// ---------------------------------------------------------------------------
// Kernel C: per (b, block i, block j):
//   Wd[e,o] = sum_f W_base[(e*256+o)*8+f] * w8[f]   (built on the fly)
//   out[e, p*8+q] = sum_o Wd[e,o] * y[b,o,i*8+p,j*8+q]
// ypatch staged into LDS by the Tensor Data Mover (3D tile: 8 x 8 x 256,
// strides 128 / 16384, LDS padding 8 DWORDs every 64 DWORDs -> pitch 72).
// 256 threads = 8 waves; fp32 WMMA 16x16x4, K-loop of 64 x 8 WMMAs.
// ---------------------------------------------------------------------------
#define YP 72  // LDS pitch: 2*72 mod 64 = 16 -> lane-halves hit disjoint banks

__global__ __launch_bounds__(256) void k_main(const float* __restrict__ y,
                                              const float* __restrict__ Wb,
                                              const float* __restrict__ wdec,
                                              float* __restrict__ out) {
  const int j = blockIdx.x;  // 0..15
  const int i = blockIdx.y;  // 0..15
  const int b = blockIdx.z;  // 0..1

  const int tid  = threadIdx.x;
  const int wv   = tid >> 5;        // wave 0..7
  const int lane = tid & 31;
  const int half = lane >> 4;       // lane half selects K pair
  const int m    = lane & 15;       // row/col within 16

  __shared__ float ylds[256 * YP];  // [O=256][PQ=64] padded, 72 KB

  const float* ybase =
      y + (size_t)b * 256 * 16384 + (size_t)(i * 8) * 128 + j * 8;

#if __has_builtin(__builtin_amdgcn_tensor_load_to_lds)
  // ---- TDM: one wave issues the whole 64KB-tile descriptor ----
  if (wv == 0) {
    const unsigned long long ga = (unsigned long long)(size_t)ybase;
    const unsigned lb = (unsigned)(size_t)&ylds[0];  // flat low 32 = LDS addr
    u32x4 g0;
    g0[0] = 1u;                                   // count = 1 valid descriptor
    g0[1] = lb;                                   // lds_addr
    g0[2] = (unsigned)(ga & 0xFFFFFFFFu);         // global_addr[31:0]
    g0[3] = (unsigned)((ga >> 32) & 0x01FFFFFFu)  // global_addr[56:32]
            | (2u << 30);                         // type = 2 ("image")
    i32x8 g1;
    g1[0] = (int)((2u << 16)     // data_size = 4B
                  | (1u << 20)   // pad_enable
                  | (5u << 22)   // pad_interval: every 64 DWORDs
                  | (7u << 25)); // pad_amount: 8 DWORDs  -> pitch 72 floats
    g1[1] = (int)(128u << 16);   // tensor_dim0 = 128 (w extent)
    g1[2] = (int)(128u << 16);   // tensor_dim1 = 128 (h extent)
    g1[3] = (int)(8u << 16);     // tile_dim0 = 8   (q, contiguous)
    g1[4] = (int)(8u | (256u << 16));  // tile_dim1 = 8 (p), tile_dim2 = 256 (o)
    g1[5] = 128;                 // tensor_dim0_stride = 128 (row pitch)
    g1[6] = (int)(16384u << 16); // tensor_dim1_stride = 16384 (channel pitch)
    g1[7] = 0;
    i32x4 g2;
    g2[0] = 256;                 // tensor_dim2 = 256 (channels)
    g2[1] = 0; g2[2] = 0; g2[3] = 0;
    i32x4 g3;
    g3[0] = 0; g3[1] = 0; g3[2] = 0; g3[3] = 0;
    i32x8 g4;                    // unused 5th VADDR group (ISA: set to NULL)
    g4[0] = 0; g4[1] = 0; g4[2] = 0; g4[3] = 0;
    g4[4] = 0; g4[5] = 0; g4[6] = 0; g4[7] = 0;
    __builtin_amdgcn_tensor_load_to_lds(g0, g1, g2, g3, g4, 0);
    __builtin_amdgcn_s_wait_tensorcnt(0);
  }
#else
  // ---- fallback: manual staged copy, row r -> (o = r>>3, p = r&7) ----
#pragma unroll
  for (int r = tid; r < 2048; r += 256) {
    int o = r >> 3, p = r & 7;
    const float4* g = (const float4*)(ybase + (size_t)o * 16384 + p * 128);
    float4 a0 = g[0], a1 = g[1];
    float* d = &ylds[o * YP + p * 8];
    *(float4*)(d)     = a0;
    *(float4*)(d + 4) = a1;
  }
#endif

  // ---- per-block filter coefficients (broadcast through cache) ----
  const float* wd = wdec + (size_t)b * 2048 + i * 16 + j;
  float4 w8a, w8b;
  w8a.x = wd[0 * 256]; w8a.y = wd[1 * 256]; w8a.z = wd[2 * 256]; w8a.w = wd[3 * 256];
  w8b.x = wd[4 * 256]; w8b.y = wd[5 * 256]; w8b.z = wd[6 * 256]; w8b.w = wd[7 * 256];

  __syncthreads();

  v8f acc[2][4];
#pragma unroll
  for (int et = 0; et < 2; ++et)
#pragma unroll
    for (int nt = 0; nt < 4; ++nt) acc[et][nt] = (v8f)(0.0f);

  const int e_lane = wv * 32 + m;  // A-row for this lane (M dim), et adds 16

  for (int ko = 0; ko < 256; ko += 4) {
    const int k0 = ko + 2 * half;  // this lane-half's K pair: {k0, k0+1}

    // Build A fragments: A[et] = (Wd[e, k0], Wd[e, k0+1]); rows k0,k0+1 of
    // W_base are 16 contiguous floats (64B) -> 4 x b128 loads, 16 FMA.
    v2f A[2];
#pragma unroll
    for (int et = 0; et < 2; ++et) {
      const int e = e_lane + et * 16;
      const float4* wrow = (const float4*)(Wb + ((size_t)(e * 256 + k0)) * 8);
      float4 r0 = wrow[0], r1 = wrow[1], r2 = wrow[2], r3 = wrow[3];
      float a0 = r0.x * w8a.x + r0.y * w8a.y + r0.z * w8a.z + r0.w * w8a.w +
                 r1.x * w8b.x + r1.y * w8b.y + r1.z * w8b.z + r1.w * w8b.w;
      float a1 = r2.x * w8a.x + r2.y * w8a.y + r2.z * w8a.z + r2.w * w8a.w +
                 r3.x * w8b.x + r3.y * w8b.y + r3.z * w8b.z + r3.w * w8b.w;
      A[et].x = a0;
      A[et].y = a1;
    }

#pragma unroll
    for (int nt = 0; nt < 4; ++nt) {
      v2f Bf;
      Bf.x = ylds[k0 * YP + nt * 16 + m];
      Bf.y = ylds[(k0 + 1) * YP + nt * 16 + m];
      acc[0][nt] = __builtin_amdgcn_wmma_f32_16x16x4_f32(
          false, A[0], false, Bf, (short)0, acc[0][nt], false, false);
      acc[1][nt] = __builtin_amdgcn_wmma_f32_16x16x4_f32(
          false, A[1], false, Bf, (short)0, acc[1][nt], false, false);
    }
  }

  // ---- epilogue: C/D layout -> out[b, e, i*8+p, j*8+q] ----
  float* obase = out + (size_t)b * 256 * 16384 + (size_t)(i * 8) * 128 + j * 8;
#pragma unroll
  for (int et = 0; et < 2; ++et) {
#pragma unroll
    for (int nt = 0; nt < 4; ++nt) {
      const int n = nt * 16 + m;
      const int p = n >> 3, q = n & 7;
#pragma unroll
      for (int r = 0; r < 8; ++r) {
        const int e = wv * 32 + et * 16 + r + 8 * half;
        obase[(size_t)e * 16384 + p * 128 + q] = acc[et][nt][r];
      }
    }
  }
}

// ---------------------------------------------------------------------------
extern "C" void kernel_launch(void* const* d_in, const int* in_sizes, int n_in,
                              void* d_out, int out_size, void* d_ws,
                              size_t ws_size, hipStream_t stream) {
  const float* x       = (const float*)d_in[0];
  const float* y       = (const float*)d_in[1];
  const float* conv1_w = (const float*)d_in[2];
  const float* conv1_b = (const float*)d_in[3];
  const float* dw_w    = (const float*)d_in[4];
  const float* dw_b    = (const float*)d_in[5];
  const float* bn_g    = (const float*)d_in[6];
  const float* bn_b    = (const float*)d_in[7];
  const float* bn_m    = (const float*)d_in[8];
  const float* bn_v    = (const float*)d_in[9];
  const float* sr1s    = (const float*)d_in[10];
  const float* sr1b    = (const float*)d_in[11];
  const float* ce_w    = (const float*)d_in[12];
  const float* ce_b    = (const float*)d_in[13];
  const float* sr2s    = (const float*)d_in[14];
  const float* sr2b    = (const float*)d_in[15];
  const float* W_base  = (const float*)d_in[16];

  float* t    = (float*)d_ws;       // 2*128*256 = 65536 floats
  float* wdec = t + 65536;          // 2*8*256   =  4096 floats
  float* out  = (float*)d_out;

  k_conv1<<<dim3(256, 2), 128, 0, stream>>>(x, conv1_w, conv1_b, t);
  k_dw_ce<<<dim3(256, 2), 128, 0, stream>>>(t, dw_w, dw_b, bn_g, bn_b, bn_m,
                                            bn_v, sr1s, sr1b, ce_w, ce_b, sr2s,
                                            sr2b, wdec);
  k_main<<<dim3(16, 16, 2), 256, 0, stream>>>(y, W_base, wdec, out);
}